// QuantumNeuralNetwork_74500502716749
// MI455X (gfx1250) — compile-verified
//
#include <hip/hip_runtime.h>
#include <math.h>

#define NQ 12
#define DEPTH 3
#define BATCH 512
#define EPSF 1e-5f
#define KC 32

typedef float v2f __attribute__((ext_vector_type(2)));
typedef float v8f __attribute__((ext_vector_type(8)));

// ---------------------------------------------------------------------------
// Small GEMM: one wave per 16x16 tile. Y[M x ldy] = X[M x K] @ W^T + bias.
// W must have >= gridDim.y*16 valid rows (pre-padded if needed) so the inner
// loop has NO guards; only the epilogue store checks col < Nreal.
// ---------------------------------------------------------------------------
__global__ __launch_bounds__(32)
void gemm_small(const float* __restrict__ X, const float* __restrict__ W,
                const float* __restrict__ bias, float* __restrict__ Y,
                int K, int Nreal, int ldy) {
  const int lane = threadIdx.x & 31;
  const int half = lane >> 4;
  const int l16  = lane & 15;
  const int tm = blockIdx.x * 16;
  const int tn = blockIdx.y * 16;
  const int col = tn + l16;

  const float* xptr = X + (size_t)(tm + l16) * K;
  const float* wptr = W + (size_t)col * K;

  v8f acc = {};
  for (int k = 0; k < K; k += 4) {
    const int ka = k + half * 2;
    const float2 af = *(const float2*)&xptr[ka];
    const float2 bf = *(const float2*)&wptr[ka];
    v2f a, b;
    a.x = af.x; a.y = af.y;
    b.x = bf.x; b.y = bf.y;
    acc = __builtin_amdgcn_wmma_f32_16x16x4_f32(false, a, false, b,
                                                (short)0, acc, false, false);
  }

  if (col < Nreal) {
    const float bb = bias[col];
#pragma unroll
    for (int r = 0; r < 8; ++r)
      Y[(size_t)(tm + r + half * 8) * ldy + col] = acc[r] + bb;
  }
}

// ---------------------------------------------------------------------------
// Big GEMM: 8 waves/block, block tile 16(M) x 128(N). A chunk (16 x KC) staged
// in LDS (stride 34 floats to spread banks); each wave owns one 16x16 N-tile.
// Requires: K % KC == 0, N % 128 == 0.
// ---------------------------------------------------------------------------
__global__ __launch_bounds__(256)
void gemm_big(const float* __restrict__ X, const float* __restrict__ W,
              const float* __restrict__ bias, float* __restrict__ Y,
              int K, int N) {
  __shared__ float As[16 * 34];
  const int tid  = threadIdx.x;
  const int lane = tid & 31;
  const int wave = tid >> 5;
  const int half = lane >> 4;
  const int l16  = lane & 15;
  const int tm = blockIdx.x * 16;
  const int tn = blockIdx.y * 128 + wave * 16;
  const int col = tn + l16;

  const int srow = tid >> 4;        // 0..15
  const int skk  = (tid & 15) * 2;  // 0,2,..,30

  const float* xrow = X + (size_t)(tm + srow) * K + skk;
  const float* wptr = W + (size_t)col * K;

  v8f acc = {};
  for (int k0 = 0; k0 < K; k0 += KC) {
    const float2 av = *(const float2*)&xrow[k0];
    if (k0 + KC < K) {
      __builtin_prefetch(&wptr[k0 + KC], 0, 3);
      __builtin_prefetch(&xrow[k0 + KC], 0, 3);
    }
    __syncthreads();                       // previous chunk fully consumed
    *(float2*)&As[srow * 34 + skk] = av;
    __syncthreads();
#pragma unroll
    for (int k = 0; k < KC; k += 4) {
      const int ka = k + half * 2;
      const float2 af = *(const float2*)&As[l16 * 34 + ka];
      const float2 bf = *(const float2*)&wptr[k0 + ka];
      v2f a, b;
      a.x = af.x; a.y = af.y;
      b.x = bf.x; b.y = bf.y;
      acc = __builtin_amdgcn_wmma_f32_16x16x4_f32(false, a, false, b,
                                                  (short)0, acc, false, false);
    }
  }

  const float bb = bias[col];
#pragma unroll
  for (int r = 0; r < 8; ++r)
    Y[(size_t)(tm + r + half * 8) * N + col] = acc[r] + bb;
}

// ---------------------------------------------------------------------------
// Pad 12xK weights -> 16xK (zeros in rows 12..15) and 12 -> 16 bias.
// ---------------------------------------------------------------------------
__global__ __launch_bounds__(256)
void padw_k(const float* __restrict__ W, const float* __restrict__ b,
            float* __restrict__ Wp, float* __restrict__ bp, int K) {
  const int idx = blockIdx.x * 256 + threadIdx.x;
  const int total = 16 * K;
  if (idx < total) {
    const int r = idx / K;
    Wp[idx] = (r < NQ) ? W[idx] : 0.f;
  }
  if (idx < 16) bp[idx] = (idx < NQ) ? b[idx] : 0.f;
}

// ---------------------------------------------------------------------------
// Precompute U = Rz*Ry*Rx per (d,q) and CRY cos/sin per (d,i).
// umat: DEPTH*NQ*8 floats (u00,u01,u10,u11 as re,im); crys: DEPTH*(NQ-1)*2.
// ---------------------------------------------------------------------------
__device__ __forceinline__ float2 cmulf(float2 a, float2 b) {
  return make_float2(a.x * b.x - a.y * b.y, a.x * b.y + a.y * b.x);
}

__global__ __launch_bounds__(64)
void gateprep_k(const float* __restrict__ rot, const float* __restrict__ ent,
                float* __restrict__ umat, float* __restrict__ crys) {
  const int t = threadIdx.x;
  if (t < DEPTH * NQ) {
    const float hx = 0.5f * rot[t * 3 + 0];
    const float hy = 0.5f * rot[t * 3 + 1];
    const float hz = 0.5f * rot[t * 3 + 2];
    const float cx = cosf(hx), sx = sinf(hx);
    const float cy = cosf(hy), sy = sinf(hy);
    const float cz = cosf(hz), sz = sinf(hz);
    // M = Ry*Rx
    const float2 m00 = make_float2(cy * cx,  sy * sx);
    const float2 m01 = make_float2(-sy * cx, -cy * sx);
    const float2 m10 = make_float2(sy * cx,  -cy * sx);
    const float2 m11 = make_float2(cy * cx,  -sy * sx);
    // U = Rz*M
    const float2 e0 = make_float2(cz, -sz);
    const float2 e1 = make_float2(cz,  sz);
    const float2 u00 = cmulf(e0, m00), u01 = cmulf(e0, m01);
    const float2 u10 = cmulf(e1, m10), u11 = cmulf(e1, m11);
    float* o = umat + t * 8;
    o[0] = u00.x; o[1] = u00.y; o[2] = u01.x; o[3] = u01.y;
    o[4] = u10.x; o[5] = u10.y; o[6] = u11.x; o[7] = u11.y;
  }
  if (t < DEPTH * (NQ - 1)) {
    const float th = 0.5f * ent[t];
    crys[t * 2 + 0] = cosf(th);
    crys[t * 2 + 1] = sinf(th);
  }
}

// ---------------------------------------------------------------------------
// Column mean/var over batch (population variance), one block per column.
// ---------------------------------------------------------------------------
__global__ __launch_bounds__(256)
void colstats_k(const float* __restrict__ Y, float* __restrict__ mu,
                float* __restrict__ var, int B, int N) {
  const int j = blockIdx.x;
  const int tid = threadIdx.x;
  float s1 = 0.f, s2 = 0.f;
  for (int i = tid; i < B; i += 256) {
    const float v = Y[(size_t)i * N + j];
    s1 += v; s2 += v * v;
  }
  __shared__ float r1[256], r2[256];
  r1[tid] = s1; r2[tid] = s2;
  __syncthreads();
  for (int off = 128; off > 0; off >>= 1) {
    if (tid < off) { r1[tid] += r1[tid + off]; r2[tid] += r2[tid + off]; }
    __syncthreads();
  }
  if (tid == 0) {
    const float m = r1[0] / (float)B;
    mu[j] = m;
    var[j] = r2[0] / (float)B - m * m;
  }
}

__global__ __launch_bounds__(256)
void bn_apply_k(const float* __restrict__ Y, const float* __restrict__ mu,
                const float* __restrict__ var, const float* __restrict__ g,
                const float* __restrict__ beta, float* __restrict__ out,
                int total, int N) {
  const int idx = blockIdx.x * 256 + threadIdx.x;
  if (idx >= total) return;
  const int j = idx % N;
  out[idx] = (Y[idx] - mu[j]) * rsqrtf(var[j] + EPSF) * g[j] + beta[j];
}

// ---------------------------------------------------------------------------
// BN + tanh + row-normalize + embedding-angle prep.
// ang = 2*asin(|a|) -> cos(ang/2)=sqrt(1-a^2), sin(ang/2)=|a|.
// ---------------------------------------------------------------------------
__global__ __launch_bounds__(64)
void angles_k(const float* __restrict__ Y0, const float* __restrict__ mu,
              const float* __restrict__ var, const float* __restrict__ g,
              const float* __restrict__ bi, float* __restrict__ ang, int B) {
  const int b = blockIdx.x * 64 + threadIdx.x;
  if (b >= B) return;
  float e[NQ];
  float n2 = 0.f;
#pragma unroll
  for (int q = 0; q < NQ; ++q) {
    const float v = (Y0[b * NQ + q] - mu[q]) * rsqrtf(var[q] + EPSF) * g[q] + bi[q];
    e[q] = tanhf(v);
    n2 += e[q] * e[q];
  }
  const float nrm = sqrtf(n2);
  const float inv = (nrm > 0.f) ? (1.f / nrm) : 1.f;
#pragma unroll
  for (int q = 0; q < NQ; ++q) {
    const float a  = e[q] * inv;
    const float aa = fminf(fabsf(a), 1.f);
    ang[b * (NQ * 3) + q * 3 + 0] = sqrtf(fmaxf(1.f - aa * aa, 0.f));
    ang[b * (NQ * 3) + q * 3 + 1] = aa;
    ang[b * (NQ * 3) + q * 3 + 2] = (a < 0.f) ? -1.f : 1.f;
  }
}

// ---------------------------------------------------------------------------
// 12-qubit state-vector simulation in LDS (4096 complex = 32KB per block).
// One block per batch row. Qubit q lives at index bit (NQ-1-q).
// ---------------------------------------------------------------------------
__global__ __launch_bounds__(256)
void qsim_k(const float* __restrict__ ang,   // B x NQ x 3
            const float* __restrict__ umat,  // DEPTH*NQ*8
            const float* __restrict__ crys,  // DEPTH*(NQ-1)*2
            float* __restrict__ m) {         // B x NQ
  __shared__ float2 st[1 << NQ];
  __shared__ float probs[NQ];
  const int b = blockIdx.x;
  const int tid = threadIdx.x;

  for (int i = tid; i < (1 << NQ); i += 256) st[i] = make_float2(0.f, 0.f);
  if (tid < NQ) probs[tid] = 0.f;
  __syncthreads();
  if (tid == 0) st[0] = make_float2(1.f, 0.f);
  __syncthreads();

  // Embedding gates: U = [[c, -s],[sg*s, sg*c]] (real)
  for (int q = 0; q < NQ; ++q) {
    const float c  = ang[b * (NQ * 3) + q * 3 + 0];
    const float si = ang[b * (NQ * 3) + q * 3 + 1];
    const float sg = ang[b * (NQ * 3) + q * 3 + 2];
    const int pos = NQ - 1 - q;
    const int str = 1 << pos;
    for (int t = tid; t < (1 << (NQ - 1)); t += 256) {
      const int i0 = ((t >> pos) << (pos + 1)) | (t & (str - 1));
      const int i1 = i0 + str;
      const float2 a0 = st[i0], a1 = st[i1];
      st[i0] = make_float2(c * a0.x - si * a1.x, c * a0.y - si * a1.y);
      st[i1] = make_float2(sg * (si * a0.x + c * a1.x), sg * (si * a0.y + c * a1.y));
    }
    __syncthreads();
  }

  for (int d = 0; d < DEPTH; ++d) {
    for (int q = 0; q < NQ; ++q) {
      const float* u = umat + (d * NQ + q) * 8;
      const float2 u00 = make_float2(u[0], u[1]);
      const float2 u01 = make_float2(u[2], u[3]);
      const float2 u10 = make_float2(u[4], u[5]);
      const float2 u11 = make_float2(u[6], u[7]);
      const int pos = NQ - 1 - q;
      const int str = 1 << pos;
      for (int t = tid; t < (1 << (NQ - 1)); t += 256) {
        const int i0 = ((t >> pos) << (pos + 1)) | (t & (str - 1));
        const int i1 = i0 + str;
        const float2 a0 = st[i0], a1 = st[i1];
        float2 n0 = cmulf(u00, a0), n1 = cmulf(u10, a0);
        const float2 t0 = cmulf(u01, a1), t1 = cmulf(u11, a1);
        n0.x += t0.x; n0.y += t0.y;
        n1.x += t1.x; n1.y += t1.y;
        st[i0] = n0; st[i1] = n1;
      }
      __syncthreads();
    }
    for (int i = 0; i < NQ - 1; ++i) {
      const float co = crys[(d * (NQ - 1) + i) * 2 + 0];
      const float si = crys[(d * (NQ - 1) + i) * 2 + 1];
      const int tgt = NQ - 2 - i;          // target bit; control = tgt+1
      const int lowmask = (1 << tgt) - 1;
      for (int t = tid; t < (1 << (NQ - 2)); t += 256) {
        const int low = t & lowmask;
        const int high = t >> tgt;
        const int i0 = (high << (tgt + 2)) | (1 << (tgt + 1)) | low;
        const int i1 = i0 + (1 << tgt);
        const float2 a0 = st[i0], a1 = st[i1];
        st[i0] = make_float2(co * a0.x - si * a1.x, co * a0.y - si * a1.y);
        st[i1] = make_float2(si * a0.x + co * a1.x, si * a0.y + co * a1.y);
      }
      __syncthreads();
    }
  }

  float acc[NQ];
#pragma unroll
  for (int q = 0; q < NQ; ++q) acc[q] = 0.f;
  for (int i = tid; i < (1 << NQ); i += 256) {
    const float2 a = st[i];
    const float p = a.x * a.x + a.y * a.y;
#pragma unroll
    for (int q = 0; q < NQ; ++q)
      if (i & (1 << (NQ - 1 - q))) acc[q] += p;
  }
#pragma unroll
  for (int q = 0; q < NQ; ++q) atomicAdd(&probs[q], acc[q]);
  __syncthreads();
  if (tid < NQ) m[b * NQ + tid] = probs[tid];
}

// ---------------------------------------------------------------------------
// Per-row LayerNorm (+ optional ReLU), one block per row.
// ---------------------------------------------------------------------------
__global__ __launch_bounds__(256)
void rowln_k(const float* __restrict__ Y, const float* __restrict__ g,
             const float* __restrict__ beta, float* __restrict__ out,
             int N, int do_relu) {
  const int row = blockIdx.x;
  const int tid = threadIdx.x;
  const float* y = Y + (size_t)row * N;
  float s1 = 0.f, s2 = 0.f;
  for (int i = tid; i < N; i += 256) { const float v = y[i]; s1 += v; s2 += v * v; }
  __shared__ float r1[256], r2[256];
  r1[tid] = s1; r2[tid] = s2;
  __syncthreads();
  for (int off = 128; off > 0; off >>= 1) {
    if (tid < off) { r1[tid] += r1[tid + off]; r2[tid] += r2[tid + off]; }
    __syncthreads();
  }
  __shared__ float mu_s, rs_s;
  if (tid == 0) {
    const float mval = r1[0] / (float)N;
    const float vval = r2[0] / (float)N - mval * mval;
    mu_s = mval;
    rs_s = rsqrtf(vval + EPSF);
  }
  __syncthreads();
  const float mval = mu_s, rs = rs_s;
  for (int i = tid; i < N; i += 256) {
    float v = (y[i] - mval) * rs * g[i] + beta[i];
    if (do_relu) v = fmaxf(v, 0.f);
    out[(size_t)row * N + i] = v;
  }
}

// ---------------------------------------------------------------------------
// Host-side pipeline
// ---------------------------------------------------------------------------
extern "C" void kernel_launch(void* const* d_in, const int* in_sizes, int n_in,
                              void* d_out, int out_size, void* d_ws, size_t ws_size,
                              hipStream_t stream) {
  (void)in_sizes; (void)n_in; (void)out_size; (void)ws_size;
  const float* x    = (const float*)d_in[0];
  const float* qWe0 = (const float*)d_in[1];
  const float* qbe0 = (const float*)d_in[2];
  const float* qgi0 = (const float*)d_in[3];
  const float* qbi0 = (const float*)d_in[4];
  const float* rot0 = (const float*)d_in[5];
  const float* ent0 = (const float*)d_in[6];
  const float* qWd0 = (const float*)d_in[7];
  const float* qbd0 = (const float*)d_in[8];
  const float* qgo0 = (const float*)d_in[9];
  const float* qbo0 = (const float*)d_in[10];
  const float* qWe1 = (const float*)d_in[11];
  const float* qbe1 = (const float*)d_in[12];
  const float* qgi1 = (const float*)d_in[13];
  const float* qbi1 = (const float*)d_in[14];
  const float* rot1 = (const float*)d_in[15];
  const float* ent1 = (const float*)d_in[16];
  const float* qWd1 = (const float*)d_in[17];
  const float* qbd1 = (const float*)d_in[18];
  const float* qgo1 = (const float*)d_in[19];
  const float* qbo1 = (const float*)d_in[20];
  const float* W2   = (const float*)d_in[21];
  const float* b2   = (const float*)d_in[22];
  const float* ln2g = (const float*)d_in[23];
  const float* ln2b = (const float*)d_in[24];
  const float* Wo   = (const float*)d_in[25];
  const float* bo   = (const float*)d_in[26];
  const float* lnog = (const float*)d_in[27];
  const float* lnob = (const float*)d_in[28];
  float* out = (float*)d_out;

  const int B = BATCH, IN = 512, H1 = 512, H2 = 512, H3 = 1024, OUT = 256;

  float* ws = (float*)d_ws;
  float* y0    = ws;                       // B*12
  float* angb  = y0    + B * NQ;           // B*36
  float* mbuf  = angb  + B * NQ * 3;       // B*12
  float* mu    = mbuf  + B * NQ;           // 1024
  float* var   = mu    + 1024;             // 1024
  float* Wep0  = var   + 1024;             // 16*512
  float* Wep1  = Wep0  + 16 * 512;         // 16*512
  float* bep0  = Wep1  + 16 * 512;         // 32
  float* bep1  = bep0  + 32;               // 32
  float* umat0 = bep1  + 32;               // 512
  float* crys0 = umat0 + 512;              // 128
  float* umat1 = crys0 + 128;              // 512
  float* crys1 = umat1 + 512;              // 128
  float* yd    = crys1 + 128;              // B*512
  float* hbuf  = yd    + B * 512;          // B*512
  float* t3    = hbuf  + B * 512;          // B*1024
  float* h3    = t3    + B * 1024;         // B*1024
  float* t4    = h3    + B * 1024;         // B*256

  // Prep (independent of data): pad encoder weights, precompute gate matrices.
  padw_k<<<(16 * IN + 255) / 256, 256, 0, stream>>>(qWe0, qbe0, Wep0, bep0, IN);
  padw_k<<<(16 * H1 + 255) / 256, 256, 0, stream>>>(qWe1, qbe1, Wep1, bep1, H1);
  gateprep_k<<<1, 64, 0, stream>>>(rot0, ent0, umat0, crys0);
  gateprep_k<<<1, 64, 0, stream>>>(rot1, ent1, umat1, crys1);

  // ---- quantum layer 0 ----
  gemm_small<<<dim3(B / 16, 1), 32, 0, stream>>>(x, Wep0, bep0, y0, IN, NQ, NQ);
  colstats_k<<<NQ, 256, 0, stream>>>(y0, mu, var, B, NQ);
  angles_k<<<(B + 63) / 64, 64, 0, stream>>>(y0, mu, var, qgi0, qbi0, angb, B);
  qsim_k<<<B, 256, 0, stream>>>(angb, umat0, crys0, mbuf);
  gemm_small<<<dim3(B / 16, H1 / 16), 32, 0, stream>>>(mbuf, qWd0, qbd0, yd, NQ, H1, H1);
  colstats_k<<<H1, 256, 0, stream>>>(yd, mu, var, B, H1);
  bn_apply_k<<<(B * H1 + 255) / 256, 256, 0, stream>>>(yd, mu, var, qgo0, qbo0, hbuf, B * H1, H1);

  // ---- quantum layer 1 ----
  gemm_small<<<dim3(B / 16, 1), 32, 0, stream>>>(hbuf, Wep1, bep1, y0, H1, NQ, NQ);
  colstats_k<<<NQ, 256, 0, stream>>>(y0, mu, var, B, NQ);
  angles_k<<<(B + 63) / 64, 64, 0, stream>>>(y0, mu, var, qgi1, qbi1, angb, B);
  qsim_k<<<B, 256, 0, stream>>>(angb, umat1, crys1, mbuf);
  gemm_small<<<dim3(B / 16, H2 / 16), 32, 0, stream>>>(mbuf, qWd1, qbd1, yd, NQ, H2, H2);
  colstats_k<<<H2, 256, 0, stream>>>(yd, mu, var, B, H2);
  bn_apply_k<<<(B * H2 + 255) / 256, 256, 0, stream>>>(yd, mu, var, qgo1, qbo1, hbuf, B * H2, H2);

  // ---- dense head ----
  gemm_big<<<dim3(B / 16, H3 / 128), 256, 0, stream>>>(hbuf, W2, b2, t3, H2, H3);
  rowln_k<<<B, 256, 0, stream>>>(t3, ln2g, ln2b, h3, H3, 1);
  gemm_big<<<dim3(B / 16, OUT / 128), 256, 0, stream>>>(h3, Wo, bo, t4, H3, OUT);
  rowln_k<<<B, 256, 0, stream>>>(t4, lnog, lnob, out, OUT, 0);
}